// GINEConv_86277303042056
// MI455X (gfx1250) — compile-verified
//
#include <hip/hip_runtime.h>

typedef float v2f __attribute__((ext_vector_type(2)));
typedef float v4f __attribute__((ext_vector_type(4)));
typedef float v8f __attribute__((ext_vector_type(8)));

#define NN 100000   // nodes
#define NE 1600000  // edges
#define DD 128      // node feature dim
#define ED 64       // edge feature dim

// -------------------------------------------------------------------------
// Kernel 1: h = (1+eps) * nodes  (eps = 0)  -> plain vectorized copy
// -------------------------------------------------------------------------
__global__ __launch_bounds__(256) void gine_init_h(const float* __restrict__ nodes,
                                                   float* __restrict__ h) {
  size_t i = (size_t)blockIdx.x * 256 + threadIdx.x;
  size_t n4 = (size_t)NN * DD / 4;
  if (i < n4) ((v4f*)h)[i] = ((const v4f*)nodes)[i];
}

// -------------------------------------------------------------------------
// Kernel 2: per-edge  e = edges @ W_edge ; msg = e + nodes[sender] ;
//           atomicAdd(h[receiver], msg)
// 256 threads = 8 waves; each wave owns a 16-edge x 128-col tile.
// W_edge^T staged in LDS, padded stride 66 (even -> b64-aligned,
// 66 % 64 == 2 -> conflict-free column reads).
// Gathers for each N-tile are issued BEFORE the 16-WMMA chain so their
// L2 latency hides under the matrix math; row base pointers are
// precomputed once per wave so the hot loop only adds n*64B offsets.
// -------------------------------------------------------------------------
#define WE_PAD 66

__global__ __launch_bounds__(256) void gine_edge(const float* __restrict__ edges,
                                                 const int* __restrict__ senders,
                                                 const int* __restrict__ receivers,
                                                 const float* __restrict__ nodes,
                                                 const float* __restrict__ W_edge,
                                                 float* __restrict__ h) {
  __shared__ float WT[DD * WE_PAD];  // WT[col*WE_PAD + k] = W_edge[k][col]

  int tid = threadIdx.x;
  for (int idx = tid; idx < ED * DD; idx += 256) {
    int k = idx >> 7;     // / DD
    int col = idx & 127;  // % DD
    WT[col * WE_PAD + k] = W_edge[idx];
  }
  __syncthreads();

  int wave = tid >> 5;
  int lane = tid & 31;
  int e0 = (blockIdx.x * 8 + wave) * 16;
  if (e0 >= NE) return;  // wave-uniform; no syncs after this point

  int m = lane & 15;            // A: row within tile / B,C: col within tile
  int kg = (lane >> 4) << 1;    // K group: 0 or 2
  int rbase = (lane >> 4) << 3; // C rows covered: 0..7 or 8..15

  // A fragments: edges[e0+m][ks*4 + kg + {0,1}]   (16 x float2, one b64 each)
  v2f a[16];
  const float* arow = edges + (size_t)(e0 + m) * ED + kg;
#pragma unroll
  for (int ks = 0; ks < 16; ++ks) a[ks] = *(const v2f*)(arow + ks * 4);

  // per-row base pointers for gather (nodes[sender] + m) / scatter (h[recv] + m)
  const float* np[8];
  float* hp[8];
#pragma unroll
  for (int r = 0; r < 8; ++r) {
    int row = e0 + rbase + r;
    np[r] = nodes + (size_t)senders[row] * DD + m;
    hp[r] = h + (size_t)receivers[row] * DD + m;
  }

#pragma unroll 1
  for (int n = 0; n < 8; ++n) {  // 8 N-tiles of 16 cols
    int coff = n << 4;
    // issue all 8 independent gathers first; latency hides under WMMAs
    float gv[8];
#pragma unroll
    for (int r = 0; r < 8; ++r) gv[r] = np[r][coff];

    const float* bcol = &WT[(coff + m) * WE_PAD + kg];
    v8f c = {};
#pragma unroll
    for (int ks = 0; ks < 16; ++ks) {  // K = 64 -> 16 steps of 4
      v2f b = *(const v2f*)(bcol + ks * 4);
      c = __builtin_amdgcn_wmma_f32_16x16x4_f32(false, a[ks], false, b,
                                                (short)0, c, false, false);
    }
    // C layout: lane holds column coff+m, rows rbase..rbase+7 in c[0..7]
#pragma unroll
    for (int r = 0; r < 8; ++r) atomicAdd(hp[r] + coff, c[r] + gv[r]);
  }
}

// -------------------------------------------------------------------------
// Kernel 3: out = relu(h @ W1 + b1) @ W2 + b2
// 256 threads = 8 waves; each wave owns 16 node rows.
// One shared weight buffer (W1^T then W2^T) + per-wave LDS tile to
// transpose layer-1 output (C layout) back into A-fragment layout.
// LDS: 66,560 + 66,560 + 1,024 B = ~131 KB (< 320 KB / WGP on CDNA5).
// -------------------------------------------------------------------------
#define WPAD 130

__global__ __launch_bounds__(256) void gine_mlp(const float* __restrict__ h,
                                                const float* __restrict__ W1,
                                                const float* __restrict__ b1,
                                                const float* __restrict__ W2,
                                                const float* __restrict__ b2,
                                                float* __restrict__ out) {
  __shared__ float WB[DD * WPAD];       // transposed weights (W1 then W2)
  __shared__ float TS[8 * 16 * WPAD];   // per-wave 16x128 activations
  __shared__ float BIAS[2 * DD];

  int tid = threadIdx.x;
  for (int idx = tid; idx < DD * DD; idx += 256) {
    int k = idx >> 7;
    int n = idx & 127;
    WB[n * WPAD + k] = W1[idx];
  }
  if (tid < DD) {
    BIAS[tid] = b1[tid];
    BIAS[DD + tid] = b2[tid];
  }
  __syncthreads();

  int wave = tid >> 5;
  int lane = tid & 31;
  int m = lane & 15;
  int kg = (lane >> 4) << 1;
  int rbase = (lane >> 4) << 3;
  int row0 = blockIdx.x * 128 + wave * 16;
  float* tw = &TS[wave * 16 * WPAD];

  // ---- layer 1: t = relu(h @ W1 + b1) ----
  v2f a[32];
  {
    int rr = row0 + m;
    if (rr >= NN) rr = NN - 1;  // clamp; OOB rows computed but never stored
    const float* arow = h + (size_t)rr * DD + kg;
#pragma unroll
    for (int ks = 0; ks < 32; ++ks) a[ks] = *(const v2f*)(arow + ks * 4);
  }
#pragma unroll 1
  for (int n = 0; n < 8; ++n) {
    int col = (n << 4) + m;
    const float* bcol = &WB[col * WPAD + kg];
    v8f c = {};
#pragma unroll
    for (int ks = 0; ks < 32; ++ks) {  // K = 128 -> 32 steps of 4
      v2f b = *(const v2f*)(bcol + ks * 4);
      c = __builtin_amdgcn_wmma_f32_16x16x4_f32(false, a[ks], false, b,
                                                (short)0, c, false, false);
    }
    float bias = BIAS[col];
#pragma unroll
    for (int r = 0; r < 8; ++r) {
      float v = c[r] + bias;
      tw[(rbase + r) * WPAD + col] = v > 0.f ? v : 0.f;
    }
  }
  __syncthreads();

  // swap weight buffer to W2^T (TS untouched)
  for (int idx = tid; idx < DD * DD; idx += 256) {
    int k = idx >> 7;
    int n = idx & 127;
    WB[n * WPAD + k] = W2[idx];
  }
  __syncthreads();

  // ---- layer 2: out = t @ W2 + b2 ----
  v2f a2[32];
  {
    const float* arow = &tw[m * WPAD + kg];
#pragma unroll
    for (int ks = 0; ks < 32; ++ks) a2[ks] = *(const v2f*)(arow + ks * 4);
  }
#pragma unroll 1
  for (int n = 0; n < 8; ++n) {
    int col = (n << 4) + m;
    const float* bcol = &WB[col * WPAD + kg];
    v8f c = {};
#pragma unroll
    for (int ks = 0; ks < 32; ++ks) {
      v2f b = *(const v2f*)(bcol + ks * 4);
      c = __builtin_amdgcn_wmma_f32_16x16x4_f32(false, a2[ks], false, b,
                                                (short)0, c, false, false);
    }
    float bias = BIAS[DD + col];
#pragma unroll
    for (int r = 0; r < 8; ++r) {
      int row = row0 + rbase + r;
      if (row < NN) out[(size_t)row * DD + col] = c[r] + bias;
    }
  }
}

// -------------------------------------------------------------------------
extern "C" void kernel_launch(void* const* d_in, const int* in_sizes, int n_in,
                              void* d_out, int out_size, void* d_ws, size_t ws_size,
                              hipStream_t stream) {
  const float* nodes     = (const float*)d_in[0];
  const int*   senders   = (const int*)d_in[1];
  const int*   receivers = (const int*)d_in[2];
  const float* edges     = (const float*)d_in[3];
  const float* W_edge    = (const float*)d_in[4];
  const float* W1        = (const float*)d_in[5];
  const float* b1        = (const float*)d_in[6];
  const float* W2        = (const float*)d_in[7];
  const float* b2        = (const float*)d_in[8];
  float* out = (float*)d_out;
  float* h   = (float*)d_ws;  // [NN x DD] aggregation buffer (51.2 MB)

  int n4 = NN * DD / 4;
  gine_init_h<<<(n4 + 255) / 256, 256, 0, stream>>>(nodes, h);
  gine_edge<<<NE / 128, 256, 0, stream>>>(edges, senders, receivers, nodes, W_edge, h);
  gine_mlp<<<(NN + 127) / 128, 256, 0, stream>>>(h, W1, b1, W2, b2, out);
}